// Quantizer_78237124264401
// MI455X (gfx1250) — compile-verified
//
#include <hip/hip_runtime.h>
#include <hip/hip_bf16.h>

#define N_ROWS 131072
#define DIM    64
#define KCB    1024
#define DECAY  0.99f
#define EPS    1e-5f

typedef __attribute__((ext_vector_type(2))) float v2f;
typedef __attribute__((ext_vector_type(8))) float v8f;

// ---------------- codebook column norms ----------------
__global__ __launch_bounds__(256) void l2cb_kernel(const float* __restrict__ cb,
                                                   float* __restrict__ l2cb) {
    int k = blockIdx.x * 256 + threadIdx.x;
    if (k < KCB) {
        float s = 0.f;
        #pragma unroll 8
        for (int d = 0; d < DIM; ++d) { float v = cb[d * KCB + k]; s += v * v; }
        l2cb[k] = s;
    }
}

// ---------------- fused GEMM (WMMA f32) + argmin ----------------
// block = 256 threads = 8 waves; each wave owns a 16-row strip (block covers 128 rows).
// A fragments register-resident; codebook tile (64x16 = 4KB) staged in LDS per block.
__global__ __launch_bounds__(256) void argmin_kernel(const float* __restrict__ flat,
                                                     const float* __restrict__ cb,
                                                     const float* __restrict__ l2cb,
                                                     int* __restrict__ idxOut) {
    __shared__ float ldsB[DIM * 16];   // [64][16]
    __shared__ float ldsL2[16];

    const int tid  = threadIdx.x;
    const int wave = tid >> 5;
    const int lane = tid & 31;
    const int half = lane >> 4;        // 0: K' in {4k,4k+1}; 1: K' in {4k+2,4k+3}
    const int l16  = lane & 15;
    const int rowBase = blockIdx.x * 128 + wave * 16;

    // A fragments: 16 k-steps of v2f (ISA 16x4 f32 A layout)
    v2f a[16];
    const float* rowPtr = flat + (size_t)(rowBase + l16) * DIM + 2 * half;
    #pragma unroll
    for (int kk = 0; kk < 16; ++kk) {
        a[kk].x = rowPtr[4 * kk + 0];
        a[kk].y = rowPtr[4 * kk + 1];
    }

    float bestD[8];
    int   bestI[8];
    #pragma unroll
    for (int v = 0; v < 8; ++v) { bestD[v] = 3.4e38f; bestI[v] = 0; }

    for (int ct = 0; ct < KCB / 16; ++ct) {
        const int c0 = ct * 16;
        __syncthreads();   // protect prior iteration's LDS reads
        for (int e = tid; e < DIM * 16; e += 256) {
            int r = e >> 4, c = e & 15;
            ldsB[e] = cb[r * KCB + c0 + c];
            if (ct + 1 < KCB / 16)  // prefetch next tile -> global_prefetch_b8
                __builtin_prefetch(&cb[r * KCB + c0 + 16 + c], 0, 1);
        }
        if (tid < 16) ldsL2[tid] = l2cb[c0 + tid];
        __syncthreads();

        v8f acc = {};
        #pragma unroll
        for (int kk = 0; kk < 16; ++kk) {
            v2f b;
            b.x = ldsB[(4 * kk + 0 + 2 * half) * 16 + l16];
            b.y = ldsB[(4 * kk + 1 + 2 * half) * 16 + l16];
            acc = __builtin_amdgcn_wmma_f32_16x16x4_f32(
                false, a[kk], false, b, (short)0, acc, false, false);
        }

        const float l2 = ldsL2[l16];
        const int   ci = c0 + l16;
        #pragma unroll
        for (int v = 0; v < 8; ++v) {
            float dist = l2 - 2.0f * acc[v];
            if (dist < bestD[v] || (dist == bestD[v] && ci < bestI[v])) {
                bestD[v] = dist; bestI[v] = ci;
            }
        }
    }

    // reduce argmin across the 16 lanes holding one row (C layout: N = lane%16)
    #pragma unroll
    for (int m = 1; m < 16; m <<= 1) {
        #pragma unroll
        for (int v = 0; v < 8; ++v) {
            float od = __shfl_xor(bestD[v], m, 32);
            int   oi = __shfl_xor(bestI[v], m, 32);
            if (od < bestD[v] || (od == bestD[v] && oi < bestI[v])) {
                bestD[v] = od; bestI[v] = oi;
            }
        }
    }
    if (l16 == 0) {
        #pragma unroll
        for (int v = 0; v < 8; ++v)
            idxOut[rowBase + half * 8 + v] = bestI[v];   // M = v + 8*half
    }
}

// ---------------- zero accumulators ----------------
__global__ __launch_bounds__(256) void zero_kernel(float* __restrict__ counts,
                                                   float* __restrict__ isum,
                                                   float* __restrict__ loss) {
    int i = blockIdx.x * 256 + threadIdx.x;
    if (i < KCB)       counts[i] = 0.f;
    if (i < DIM * KCB) isum[i]   = 0.f;
    if (i == 0)        *loss     = 0.f;
}

// ---------------- scatter-add statistics ----------------
__global__ __launch_bounds__(256) void scatter_kernel(const float* __restrict__ flat,
                                                      const int* __restrict__ idx,
                                                      float* __restrict__ counts,
                                                      float* __restrict__ isum) {
    int g = blockIdx.x * 256 + threadIdx.x;   // g < N_ROWS*DIM
    int n = g >> 6, d = g & 63;
    int k = idx[n];
    atomicAdd(&isum[d * KCB + k], flat[g]);
    if (d == 0) atomicAdd(&counts[k], 1.0f);
}

// ---------------- EMA cluster sizes + normalization ----------------
__global__ __launch_bounds__(1024) void ema_ecs_kernel(const float* __restrict__ ecs,
                                                       const float* __restrict__ counts,
                                                       float* __restrict__ out_ecs,
                                                       float* __restrict__ cluster_size) {
    __shared__ float red[1024];
    int k = threadIdx.x;
    float ne = DECAY * ecs[k] + (1.0f - DECAY) * counts[k];
    out_ecs[k] = ne;
    red[k] = ne;
    __syncthreads();
    for (int s = 512; s > 0; s >>= 1) {
        if (k < s) red[k] += red[k + s];
        __syncthreads();
    }
    float n = red[0];
    cluster_size[k] = ((ne + EPS) / (n + (float)KCB * EPS)) * n;
}

// ---------------- EMA input sums + new codebook (+transposed copy) ----------------
__global__ __launch_bounds__(256) void ema_eis_kernel(const float* __restrict__ eis,
                                                      const float* __restrict__ isum,
                                                      const float* __restrict__ cluster_size,
                                                      float* __restrict__ out_eis,
                                                      float* __restrict__ out_cb,
                                                      float* __restrict__ cbT) {
    int i = blockIdx.x * 256 + threadIdx.x;   // i < DIM*KCB
    int k = i & (KCB - 1);
    int d = i >> 10;
    float ne = DECAY * eis[i] + (1.0f - DECAY) * isum[i];
    out_eis[i] = ne;
    float c = ne / cluster_size[k];
    out_cb[i] = c;
    cbT[k * DIM + d] = c;
}

// ---------------- gather quantized + loss ----------------
__global__ __launch_bounds__(256) void quantize_kernel(const float* __restrict__ flat,
                                                       const int* __restrict__ idx,
                                                       const float* __restrict__ cbT,
                                                       float* __restrict__ outQ,
                                                       float* __restrict__ loss) {
    __shared__ float red[256];
    int g = blockIdx.x * 256 + threadIdx.x;
    int n = g >> 6, d = g & 63;
    float q = cbT[idx[n] * DIM + d];     // coalesced: 256B contiguous per row
    float x = flat[g];
    outQ[g] = q;                          // quantized_st forward value == quantized
    float diff = q - x;
    red[threadIdx.x] = diff * diff;
    __syncthreads();
    for (int s = 128; s > 0; s >>= 1) {
        if (threadIdx.x < s) red[threadIdx.x] += red[threadIdx.x + s];
        __syncthreads();
    }
    if (threadIdx.x == 0)
        atomicAdd(loss, red[0] * (1.0f / (float)(N_ROWS * DIM)));
}

extern "C" void kernel_launch(void* const* d_in, const int* in_sizes, int n_in,
                              void* d_out, int out_size, void* d_ws, size_t ws_size,
                              hipStream_t stream) {
    const float* flat = (const float*)d_in[0];   // [B,H,W,D] -> [N, D]
    const float* cb   = (const float*)d_in[1];   // [D, K]
    const float* ecs  = (const float*)d_in[2];   // [K]
    const float* eis  = (const float*)d_in[3];   // [D, K]

    float* out    = (float*)d_out;
    float* o_q    = out;                                     // 8388608
    float* o_loss = out + 8388608;                           // 1
    float* o_cb   = out + 8388609;                           // 65536
    float* o_ecs  = out + 8388609 + 65536;                   // 1024
    float* o_eis  = out + 8388609 + 65536 + 1024;            // 65536

    char*  ws      = (char*)d_ws;
    int*   w_idx   = (int*)  (ws);                           // 131072 ints
    float* w_cnt   = (float*)(ws + 524288);                  // 1024 f
    float* w_isum  = (float*)(ws + 528384);                  // 65536 f
    float* w_csz   = (float*)(ws + 790528);                  // 1024 f
    float* w_cbT   = (float*)(ws + 794624);                  // 65536 f
    float* w_l2cb  = (float*)(ws + 1056768);                 // 1024 f

    zero_kernel   <<<256, 256, 0, stream>>>(w_cnt, w_isum, o_loss);
    l2cb_kernel   <<<KCB / 256, 256, 0, stream>>>(cb, w_l2cb);
    argmin_kernel <<<N_ROWS / 128, 256, 0, stream>>>(flat, cb, w_l2cb, w_idx);
    scatter_kernel<<<(N_ROWS * DIM) / 256, 256, 0, stream>>>(flat, w_idx, w_cnt, w_isum);
    ema_ecs_kernel<<<1, 1024, 0, stream>>>(ecs, w_cnt, o_ecs, w_csz);
    ema_eis_kernel<<<(DIM * KCB) / 256, 256, 0, stream>>>(eis, w_isum, w_csz, o_eis, o_cb, w_cbT);
    quantize_kernel<<<(N_ROWS * DIM) / 256, 256, 0, stream>>>(flat, w_idx, w_cbT, o_q, o_loss);
}